// SpatialEmbLoss_36610301231308
// MI455X (gfx1250) — compile-verified
//
#include <hip/hip_runtime.h>
#include <math.h>

// ---------------------------------------------------------------------------
// SpatialEmbLoss for MI455X (gfx1250), compile-only target.
//
// Pipeline (all L2-resident after first touch; working set ~42MB << 192MB L2):
//  1. zero_kernel     : zero workspace (determinism across graph replays)
//  2. stats_kernel    : per-(b,k) masked sums as a one-hot GEMM on
//                       V_WMMA_F32_16X16X4_F32 (f32 matrix pipe, wave32)
//  3. finalize_kernel : center / s_mean / s_exp per (b,k)
//  4. hist_kernel     : fused dist + error histogram (monotone float-bit
//                       counting sort, 2^17 bins) + var/seed partials
//  5. scan_kernel     : descending prefix over bins; Lovasz grad telescopes
//                       within equal-key runs -> exact given prefix counts
//  6. final_kernel    : assemble scalar loss
//
// Transcendentals use gfx1250-native paths: v_tanh_f32 (new on CDNA5),
// v_exp_f32 (2^x) and v_rcp_f32 -- branchless, no EXEC churn around WMMA.
// A-fragment staging LDS is padded to 16 rows (rows 5..15 zeroed once) so
// fragment reads are unconditional ds_load_b32: no saveexec in the WMMA loop.
// ---------------------------------------------------------------------------

typedef __attribute__((ext_vector_type(2))) float v2f;
typedef __attribute__((ext_vector_type(8))) float v8f;
typedef unsigned long long u64;

#define BATCH 4
#define KINST 8
#define HH 512
#define WW 1024
#define HWPIX 524288           // 512*1024
#define SHIFT 13
#define NB 131073              // (0x40000000 >> 13) + 1, e in [0,2]
#define NBP 131136             // padded to multiple of 64
#define NSEG 32                // BATCH * KINST
#define LOG2E 1.4426950408889634f

__device__ __forceinline__ void atomAddF(float* p, float v) {
  __hip_atomic_fetch_add(p, v, __ATOMIC_RELAXED, __HIP_MEMORY_SCOPE_AGENT);
}
__device__ __forceinline__ void atomAddFLds(float* p, float v) {
  __hip_atomic_fetch_add(p, v, __ATOMIC_RELAXED, __HIP_MEMORY_SCOPE_WORKGROUP);
}

// Branchless transcendentals on the CDNA5 trans pipe.
__device__ __forceinline__ float fast_exp(float x) {   // e^x
#if __has_builtin(__builtin_amdgcn_exp2f)
  return __builtin_amdgcn_exp2f(x * LOG2E);            // v_exp_f32 = 2^x
#else
  return expf(x);
#endif
}
__device__ __forceinline__ float fast_exp2(float x) {  // 2^x
#if __has_builtin(__builtin_amdgcn_exp2f)
  return __builtin_amdgcn_exp2f(x);
#else
  return exp2f(x);
#endif
}
__device__ __forceinline__ float fast_tanh(float x) {
#if __has_builtin(__builtin_amdgcn_tanhf)
  return __builtin_amdgcn_tanhf(x);                    // v_tanh_f32 (gfx1250)
#else
  return tanhf(x);
#endif
}
__device__ __forceinline__ float fast_rcp(float x) {
#if __has_builtin(__builtin_amdgcn_rcpf)
  return __builtin_amdgcn_rcpf(x);                     // v_rcp_f32
#else
  return 1.0f / x;
#endif
}
__device__ __forceinline__ float fast_sigmoid(float x) {
  return fast_rcp(1.0f + fast_exp(-x));
}

// jac "after t elements, c of them positive"; jac(0)=0 is the virtual start
// so grad_0 = jac_0 (matches concat([jac[:1], diff(jac)])).
__device__ __forceinline__ float jacF(float t, float c, float P) {
  return (t > 0.0f) ? (1.0f - (P - c) / (P + t - c)) : 0.0f;
}

// ---------------------------------------------------------------------------
__global__ void zero_kernel(unsigned* __restrict__ p, size_t nwords) {
  size_t i = (size_t)blockIdx.x * blockDim.x + threadIdx.x;
  size_t stride = (size_t)gridDim.x * blockDim.x;
  for (; i < nwords; i += stride) p[i] = 0u;
}

// ---------------------------------------------------------------------------
// stats_kernel: C[M][N] += A[M][p] * B[p][N] over pixels p, where
//   A rows (M): 0=1.0 (count), 1=x-emb, 2=y-emb, 3=sigma, 4=seed^2, 5..15=0
//   B cols (N): one-hot of instance id (0..8), cols 9..15 unused (zero)
// WMMA f32 16x16x4: A = 2 VGPR (lanes 0-15: K=0/1, lanes 16-31: K=2/3,
// M = lane&15); B/C row-striped across lanes (N = lane&15).
// Grid: (128, BATCH) x 256 threads = 1024 waves; 512 pixels/wave, exact.
__global__ void __launch_bounds__(256)
stats_kernel(const float* __restrict__ pred, const int* __restrict__ inst,
             float* __restrict__ stats) {
  const int b    = blockIdx.y;
  const int tid  = threadIdx.x;
  const int lane = tid & 31;
  const int wv   = tid >> 5;
  const int gw   = blockIdx.x * 8 + wv;              // 0..1023

  const float* p0 = pred + ((size_t)(b * 4 + 0) * 2 + 1) * HWPIX;
  const float* p1 = pred + ((size_t)(b * 4 + 1) * 2 + 1) * HWPIX;
  const float* p2 = pred + ((size_t)(b * 4 + 2) * 2 + 1) * HWPIX;
  const float* p3 = pred + ((size_t)(b * 4 + 3) * 2 + 1) * HWPIX;
  const int*   ii = inst + ((size_t)b * 2 + 1) * HWPIX;

  __shared__ float feat[8][32][16];  // per-wave staging (in-order DS per wave)
  __shared__ float red[8][32][8];    // cross-wave C reduction

  // Zero the loop-invariant A rows 5..15 once -> unconditional fragment reads.
#pragma unroll
  for (int f = 5; f < 16; ++f) feat[wv][lane][f] = 0.0f;

  v8f c = {};                        // 16x16 f32 accumulator, 8 VGPRs
  const int M  = lane & 15;
  const int kb = (lane >> 4) * 2;    // K-pair base: 0 or 2

  for (int it = 0; it < 16; ++it) {
    const int p  = (gw * 16 + it) * 32 + lane;
    const float x0 = p0[p], x1 = p1[p], x2 = p2[p], x3 = p3[p];
    const int   id = ii[p];
    const float sx = fast_tanh(x0) + (float)(p & (WW - 1)) * (2.0f / 2047.0f);
    const float sy = fast_tanh(x1) + (float)(p >> 10) * (1.0f / 1023.0f);
    const float sd = fast_sigmoid(x3);
    feat[wv][lane][0] = 1.0f;
    feat[wv][lane][1] = sx;
    feat[wv][lane][2] = sy;
    feat[wv][lane][3] = x2;
    feat[wv][lane][4] = sd * sd;     // bg seed term lands in col 0 (inst==0)

#pragma unroll
    for (int g = 0; g < 8; ++g) {    // 8 WMMAs cover the wave's 32 pixels
      const int pa = g * 4 + kb;
      v2f a, bf;
      a.x = feat[wv][pa][M];         // rows >=5 read zeros (pre-filled)
      a.y = feat[wv][pa + 1][M];
      const int iA = __shfl(id, g * 4 +     (lane >> 4), 32);
      const int iB = __shfl(id, g * 4 + 2 + (lane >> 4), 32);
      bf.x = (iA == M) ? 1.0f : 0.0f;
      bf.y = (iB == M) ? 1.0f : 0.0f;
      c = __builtin_amdgcn_wmma_f32_16x16x4_f32(
          /*neg_a=*/false, a, /*neg_b=*/false, bf,
          /*c_mod=*/(short)0, c, /*reuse_a=*/false, /*reuse_b=*/false);
    }
  }

  // Cross-wave reduction: C element (M = r + (lane<16?0:8), N = lane&15)
#pragma unroll
  for (int r = 0; r < 8; ++r) red[wv][lane][r] = c[r];
  __syncthreads();
  const int Mr = tid >> 4, Nr = tid & 15;
  const int sl = ((Mr >= 8) ? 16 : 0) + Nr;
  const int sr = Mr & 7;
  float s = 0.0f;
#pragma unroll
  for (int w = 0; w < 8; ++w) s += red[w][sl][sr];
  atomAddF(&stats[(size_t)b * 256 + tid], s);
}

// ---------------------------------------------------------------------------
__global__ void finalize_kernel(const float* __restrict__ stats,
                                float* __restrict__ derived) {
  const int t = threadIdx.x;
  if (t >= 32) return;
  const int b = t >> 3, kk = (t & 7) + 1;
  const float* S = stats + (size_t)b * 256;
  const float cnt  = S[0 * 16 + kk];
  const float safe = fmaxf(cnt, 1.0f);
  const float cx = S[1 * 16 + kk] / safe;
  const float cy = S[2 * 16 + kk] / safe;
  const float sm = S[3 * 16 + kk] / safe;
  float* D = derived + (size_t)(b * 8 + (kk - 1)) * 4;
  D[0] = cx;
  D[1] = cy;
  D[2] = fast_exp(10.0f * sm) * LOG2E;  // pre-folded: d = 2^(-r2 * D[2])
  D[3] = sm;
}

// ---------------------------------------------------------------------------
// hist_kernel: per pixel, for every k compute dist/error; counting-sort the
// error into monotone float-bit buckets (packed u64: hi=pos count, lo=all),
// accumulate per-bucket error sum, plus var/seed partials for the pixel's
// own instance. Inner loop is fma + v_exp_f32 per k, branchless.
__global__ void __launch_bounds__(256)
hist_kernel(const float* __restrict__ pred, const int* __restrict__ inst,
            const float* __restrict__ derived, u64* __restrict__ cnts,
            float* __restrict__ sums, float* __restrict__ partials) {
  const int b   = blockIdx.y;
  const int tid = threadIdx.x;
  const float* p0 = pred + ((size_t)(b * 4 + 0) * 2 + 1) * HWPIX;
  const float* p1 = pred + ((size_t)(b * 4 + 1) * 2 + 1) * HWPIX;
  const float* p2 = pred + ((size_t)(b * 4 + 2) * 2 + 1) * HWPIX;
  const float* p3 = pred + ((size_t)(b * 4 + 3) * 2 + 1) * HWPIX;
  const int*   ii = inst + ((size_t)b * 2 + 1) * HWPIX;

  __shared__ float lvar[8], lseed[8];
  if (tid < 8) { lvar[tid] = 0.0f; lseed[tid] = 0.0f; }
  __syncthreads();

  float cx[8], cy[8], sxk[8], smk[8];
#pragma unroll
  for (int k = 0; k < 8; ++k) {
    const float* D = derived + (size_t)(b * 8 + k) * 4;
    cx[k] = D[0]; cy[k] = D[1]; sxk[k] = D[2]; smk[k] = D[3];
  }

  const int stride = gridDim.x * 256;
  for (int p = blockIdx.x * 256 + tid; p < HWPIX; p += stride) {
    __builtin_prefetch(p0 + p + stride, 0, 1);     // global_prefetch_b8
    __builtin_prefetch(p3 + p + stride, 0, 1);
    const float x0 = p0[p], x1 = p1[p], sg = p2[p], x3 = p3[p];
    const int   id = ii[p];
    const float sx = fast_tanh(x0) + (float)(p & (WW - 1)) * (2.0f / 2047.0f);
    const float sy = fast_tanh(x1) + (float)(p >> 10) * (1.0f / 1023.0f);
    const float sd = fast_sigmoid(x3);

#pragma unroll
    for (int k = 0; k < 8; ++k) {
      const float dx = sx - cx[k], dy = sy - cy[k];
      const float d  = fast_exp2(-(dx * dx + dy * dy) * sxk[k]); // log2e folded
      const bool  m  = (id == k + 1);
      const float e  = m ? (2.0f - 2.0f * d) : (2.0f * d);   // e in [0,2]
      unsigned bkt = __float_as_uint(e) >> SHIFT;             // monotone
      if (bkt > NB - 1) bkt = NB - 1;
      const size_t seg = (size_t)(b * 8 + k) * NBP;
      atomicAdd(&cnts[seg + bkt], (m ? ((u64)1 << 32) : (u64)0) | 1ull);
      atomAddF(&sums[seg + bkt], e);
      if (m) {
        const float dv = sg - smk[k];
        const float de = sd - d;
        atomAddFLds(&lvar[k],  dv * dv);
        atomAddFLds(&lseed[k], de * de);
      }
    }
  }
  __syncthreads();
  if (tid < 8) {
    atomAddF(&partials[(size_t)(b * 8 + tid) * 2 + 0], lvar[tid]);
    atomAddF(&partials[(size_t)(b * 8 + tid) * 2 + 1], lseed[tid]);
  }
}

// ---------------------------------------------------------------------------
// scan_kernel: one block per (b,k). Descending prefix sums (n_all, n_pos)
// over buckets; each bucket's Lovasz contribution = mean_e * (jac(after) -
// jac(before)) -- independent given the prefixes, so only a scan is needed.
__global__ void __launch_bounds__(256)
scan_kernel(const u64* __restrict__ cnts, const float* __restrict__ sums,
            const float* __restrict__ stats, float* __restrict__ instl) {
  const int seg = blockIdx.x;             // 0..31
  const int b = seg >> 3, kk = (seg & 7) + 1;
  const float P = stats[(size_t)b * 256 + kk];   // total positives (exact)
  const u64*   C = cnts + (size_t)seg * NBP;
  const float* E = sums + (size_t)seg * NBP;

  __shared__ unsigned snl[256], spl[256];
  __shared__ float redl[256];
  const int tid = threadIdx.x;
  unsigned carN = 0, carP = 0;
  float acc = 0.0f;

  const int nchunk = (NB + 255) / 256;
  for (int chk = 0; chk < nchunk; ++chk) {
    const int idx = chk * 256 + tid;      // descending position
    const int bi  = NB - 1 - idx;
    unsigned n = 0, pp = 0; float se = 0.0f;
    if (bi >= 0) {
      const u64 v = C[bi];
      n  = (unsigned)(v & 0xffffffffull);
      pp = (unsigned)(v >> 32);
      se = E[bi];
    }
    snl[tid] = n; spl[tid] = pp;
    __syncthreads();
    for (int off = 1; off < 256; off <<= 1) {     // Hillis-Steele inclusive
      const unsigned an = (tid >= off) ? snl[tid - off] : 0u;
      const unsigned ap = (tid >= off) ? spl[tid - off] : 0u;
      __syncthreads();
      snl[tid] += an; spl[tid] += ap;
      __syncthreads();
    }
    const unsigned I1 = carN + snl[tid];
    const unsigned C1 = carP + spl[tid];
    const unsigned totN = snl[255], totP = spl[255];
    if (n > 0) {
      const float t1 = (float)I1,        c1 = (float)C1;         // <2^24 exact
      const float t0 = (float)(I1 - n),  c0 = (float)(C1 - pp);
      const float ebar = se / (float)n;                          // >= 0 (relu)
      acc += ebar * (jacF(t1, c1, P) - jacF(t0, c0, P));
    }
    carN += totN; carP += totP;
    __syncthreads();
  }

  redl[tid] = acc; __syncthreads();
  for (int off = 128; off; off >>= 1) {
    if (tid < off) redl[tid] += redl[tid + off];
    __syncthreads();
  }
  if (tid == 0) instl[seg] = redl[0];
}

// ---------------------------------------------------------------------------
__global__ void final_kernel(const float* __restrict__ stats,
                             const float* __restrict__ partials,
                             const float* __restrict__ instl,
                             float* __restrict__ out) {
  if (threadIdx.x != 0) return;
  float tot = 0.0f;
  for (int b = 0; b < BATCH; ++b) {
    const float* S = stats + (size_t)b * 256;
    const float bg = S[4 * 16 + 0];           // sum(seed^2 where inst==0)
    float il = 0.0f, vl = 0.0f, sl = 0.0f;
    for (int k = 1; k <= KINST; ++k) {
      const float cnt = S[k];                 // row 0 = counts
      if (cnt > 0.0f) {
        const float safe = fmaxf(cnt, 1.0f);
        il += instl[b * 8 + k - 1];
        vl += partials[(size_t)(b * 8 + k - 1) * 2 + 0] / safe;
        sl += partials[(size_t)(b * 8 + k - 1) * 2 + 1];
      }
    }
    tot += 1.0f * il + 10.0f * vl + 1.0f * ((bg + sl) / (float)HWPIX);
  }
  out[0] = tot / (float)(BATCH - 1);
}

// ---------------------------------------------------------------------------
extern "C" void kernel_launch(void* const* d_in, const int* in_sizes, int n_in,
                              void* d_out, int out_size, void* d_ws,
                              size_t ws_size, hipStream_t stream) {
  (void)in_sizes; (void)n_in; (void)out_size; (void)ws_size;
  const float* pred = (const float*)d_in[0];
  const int*   inst = (const int*)d_in[1];   // d_in[2] (labels) unused
  float* out = (float*)d_out;

  // Workspace layout (~50.4 MB total)
  char* ws = (char*)d_ws;
  u64*   cnts     = (u64*)ws;                                   // 32*NBP u64
  float* sums     = (float*)(ws + (size_t)NSEG * NBP * 8);      // 32*NBP f32
  float* stats    = sums + (size_t)NSEG * NBP;                  // 4*256
  float* derived  = stats + BATCH * 256;                        // 4*8*4
  float* partials = derived + BATCH * 8 * 4;                    // 4*8*2
  float* instl    = partials + BATCH * 8 * 2;                   // 32

  const size_t nwords =
      ((size_t)NSEG * NBP * 8 + (size_t)NSEG * NBP * 4 +
       (size_t)(BATCH * 256 + BATCH * 32 + BATCH * 16 + NSEG) * 4) / 4;

  zero_kernel<<<2048, 256, 0, stream>>>((unsigned*)ws, nwords);
  stats_kernel<<<dim3(128, BATCH), 256, 0, stream>>>(pred, inst, stats);
  finalize_kernel<<<1, 32, 0, stream>>>(stats, derived);
  hist_kernel<<<dim3(512, BATCH), 256, 0, stream>>>(pred, inst, derived, cnts,
                                                    sums, partials);
  scan_kernel<<<NSEG, 256, 0, stream>>>(cnts, sums, stats, instl);
  final_kernel<<<1, 32, 0, stream>>>(stats, partials, instl, out);
}